// MidMaxPooling2D_47193100649159
// MI455X (gfx1250) — compile-verified
//
#include <hip/hip_runtime.h>

// MidMaxPooling2D on [16,256,256,64] f32, pool=2, ALPHA=0.5.
// out[b,ho,wo,c] = 0.5*max4 + 0.5*max(second_smallest_of_4, 0)
// Pure streaming op: ~336 MB traffic -> ~14.4 us at 23.3 TB/s. Strategy:
// b128 per-lane loads of the 4 window taps (each tap is 256B contiguous per
// 16 lanes), staged through LDS with gfx1250 async global->LDS DMA
// (ASYNCcnt-tracked), non-temporal b128 stores (single-use data).

typedef float __attribute__((ext_vector_type(4))) f4;
typedef int   __attribute__((ext_vector_type(4))) i4;  // builtin's pointee type

#define AS_GLOBAL __attribute__((address_space(1)))
#define AS_LDS    __attribute__((address_space(3)))

#if defined(__HIP_DEVICE_COMPILE__) && \
    __has_builtin(__builtin_amdgcn_global_load_async_to_lds_b128)
#define USE_ASYNC_LDS 1
#else
#define USE_ASYNC_LDS 0
#endif

static __device__ __forceinline__ f4 vmin4(f4 a, f4 b) {
  f4 r;
#pragma unroll
  for (int i = 0; i < 4; ++i) r[i] = fminf(a[i], b[i]);
  return r;
}
static __device__ __forceinline__ f4 vmax4(f4 a, f4 b) {
  f4 r;
#pragma unroll
  for (int i = 0; i < 4; ++i) r[i] = fmaxf(a[i], b[i]);
  return r;
}

__global__ __launch_bounds__(256) void midmax_pool_kernel(
    const float* __restrict__ x, float* __restrict__ out, int n4) {
  // Geometry (compile-time): B=16, H=W=256, C=64 -> Ho=Wo=128, C4=16 float4s.
  constexpr int C4  = 16;        // 64 channels / 4
  constexpr int ROW = 256 * C4;  // one input row = W*C/4 = 4096 float4

  const int gid = blockIdx.x * 256 + threadIdx.x;
  if (gid >= n4) return;

  // gid -> (b, ho, wo, c4); output is NHWC-contiguous in float4 units.
  const int c4 = gid & (C4 - 1);        // bits 0..3
  const int wo = (gid >> 4) & 127;      // bits 4..10
  const int ho = (gid >> 11) & 127;     // bits 11..17
  const int b  = gid >> 18;             // bits 18..21

  const f4* __restrict__ in4 = (const f4*)x;
  const size_t base =
      ((((size_t)b * 256 + 2 * (size_t)ho) * 256) + 2 * (size_t)wo) * C4 + c4;
  // Window taps (float4 units): (h,w), (h,w+1), (h+1,w), (h+1,w+1)
  const size_t i00 = base;
  const size_t i01 = base + C4;
  const size_t i10 = base + ROW;
  const size_t i11 = base + ROW + C4;

  f4 a, bb, c, d;
#if USE_ASYNC_LDS
  // Stage the 4 taps via async global->LDS DMA (ASYNCcnt path). Layout
  // st[k][tid]: per b128 op, consecutive lanes hit consecutive 16B -> no bank
  // conflicts on either the DMA writes or the ds_load_b128 reads. Each thread
  // reads only its own slots, so a wave-local s_wait_asynccnt(0) suffices —
  // no workgroup barrier.
  __shared__ f4 st[4][256];
  const int tid = threadIdx.x;
  __builtin_amdgcn_global_load_async_to_lds_b128(
      (AS_GLOBAL i4*)(in4 + i00), (AS_LDS i4*)&st[0][tid], 0, 0);
  __builtin_amdgcn_global_load_async_to_lds_b128(
      (AS_GLOBAL i4*)(in4 + i01), (AS_LDS i4*)&st[1][tid], 0, 0);
  __builtin_amdgcn_global_load_async_to_lds_b128(
      (AS_GLOBAL i4*)(in4 + i10), (AS_LDS i4*)&st[2][tid], 0, 0);
  __builtin_amdgcn_global_load_async_to_lds_b128(
      (AS_GLOBAL i4*)(in4 + i11), (AS_LDS i4*)&st[3][tid], 0, 0);
#if __has_builtin(__builtin_amdgcn_s_wait_asynccnt)
  __builtin_amdgcn_s_wait_asynccnt(0);
  asm volatile("" ::: "memory");
#else
  asm volatile("s_wait_asynccnt 0x0" ::: "memory");
#endif
  a  = st[0][tid];
  bb = st[1][tid];
  c  = st[2][tid];
  d  = st[3][tid];
#else
  // Fallback: direct non-temporal b128 loads (data is single-use; TH=NT
  // keeps the 256MB stream from rinsing L2).
  a  = __builtin_nontemporal_load(in4 + i00);
  bb = __builtin_nontemporal_load(in4 + i01);
  c  = __builtin_nontemporal_load(in4 + i10);
  d  = __builtin_nontemporal_load(in4 + i11);
#endif

  // max4 and 2nd-smallest-of-4 via a min/max network (8 ops):
  //   2nd_smallest = min(max(lo1,lo2), min(hi1,hi2))
  const f4 lo1 = vmin4(a, bb), hi1 = vmax4(a, bb);
  const f4 lo2 = vmin4(c, d),  hi2 = vmax4(c, d);
  const f4 mx  = vmax4(hi1, hi2);
  const f4 ss  = vmin4(vmax4(lo1, lo2), vmin4(hi1, hi2));
  const f4 zero = (f4)(0.0f);
  const f4 mid  = vmax4(ss, zero);          // masked-window max => clamp at 0
  const f4 r    = 0.5f * (mx + mid);        // ALPHA = 0.5

  __builtin_nontemporal_store(r, (f4*)out + gid);
}

extern "C" void kernel_launch(void* const* d_in, const int* in_sizes, int n_in,
                              void* d_out, int out_size, void* d_ws,
                              size_t ws_size, hipStream_t stream) {
  (void)in_sizes; (void)n_in; (void)d_ws; (void)ws_size;
  const float* x = (const float*)d_in[0];
  float* out = (float*)d_out;
  const int n4 = out_size / 4;              // 4,194,304 float4 outputs
  const int blocks = (n4 + 255) / 256;      // 16,384 blocks of 8 wave32s
  midmax_pool_kernel<<<blocks, 256, 0, stream>>>(x, out, n4);
}